// DecoderRNN_4561255268948
// MI455X (gfx1250) — compile-verified
//
#include <hip/hip_runtime.h>
#include <hip/hip_bf16.h>

// ---------------------------------------------------------------------------
// DecoderRNN for MI455X (gfx1250, wave32, WMMA).
// Shapes: V=32000 E=H=512 L=2 B=32 S=128 LEN=128.
//   1) gather embeddings -> Xemb[4096,512]
//   2) GI0 = Xemb @ wih0^T + bih0      (batched f32-WMMA GEMM)
//   3) 128 sequential steps x 2 kernels (GRU layer0, layer1); attention is
//      off the recurrence critical path, so it is deferred and batched.
//   4) Q = H1all @ win^T ; softmax/mix ; O = tanh([mix,q] @ wout^T)
//   5) scores = O @ out_w^T + out_b    (134 GFLOP f32-WMMA GEMM, row-remapped
//      from [t*B+b] to [B,S,V])
// GEMMs use V_WMMA_F32_16X16X4_F32 with 32x32 register blocking per wave
// (4 accumulators, 1 global b64 load per WMMA -> 8 FLOP/byte from cache).
// ---------------------------------------------------------------------------

typedef __attribute__((ext_vector_type(2))) float v2f;
typedef __attribute__((ext_vector_type(8))) float v8f;

#define Bsz 32
#define Ssz 128
#define Hsz 512
#define Vsz 32000
#define LENsz 128

__device__ __forceinline__ v8f wmma4(v2f a, v2f b, v8f c) {
  // (neg_a, A, neg_b, B, c_mod, C, reuse_a, reuse_b)
  return __builtin_amdgcn_wmma_f32_16x16x4_f32(false, a, false, b, (short)0, c,
                                               false, false);
}

__device__ __forceinline__ float sigmoidf_(float x) {
  return 1.0f / (1.0f + __expf(-x));
}

// ---------------------------------------------------------------------------
// Embedding gather: X[t*B+b, :] = emb_table[text_vec[b, t], :]
// ---------------------------------------------------------------------------
__global__ void __launch_bounds__(256)
embed_gather(const int* __restrict__ text, const float* __restrict__ emb,
             float* __restrict__ X) {
  int r = blockIdx.x;            // t*B + b
  int t = r >> 5, b = r & 31;
  int idx = text[b * Ssz + t];
  const float* src = emb + (size_t)idx * Hsz;
  float* dst = X + (size_t)r * Hsz;
  for (int i = threadIdx.x; i < Hsz; i += 256) dst[i] = src[i];
}

// ---------------------------------------------------------------------------
// 16x16 f32 C/D tile store: element (v,lane) = D[v + 8*(lane>=16)][lane%16].
// remap=1: output row r = t*B+b scattered to [b, t] of a [B,S,V] tensor.
// ---------------------------------------------------------------------------
__device__ __forceinline__ void store_tile(v8f acc, int mt, int nt, int lrow,
                                           int khalf,
                                           const float* __restrict__ bias,
                                           float* __restrict__ D, int ldd,
                                           int act, int remap) {
  const int col = nt + lrow;
  const float bv = bias ? bias[col] : 0.0f;
  const int mb = mt + 8 * khalf;
#pragma unroll
  for (int v = 0; v < 8; ++v) {
    int row = mb + v;
    float val = acc[v] + bv;
    if (act == 1) val = tanhf(val);
    size_t off;
    if (remap) {               // row = t*B + b  ->  [b, t, col] in [B,S,V]
      int t = row >> 5;
      int bb = row & 31;
      off = ((size_t)bb * Ssz + t) * Vsz + col;
    } else {
      off = (size_t)row * ldd + col;
    }
    D[off] = val;
  }
}

// ---------------------------------------------------------------------------
// Generic f32 WMMA GEMM: D[M,N] = act(A[M,K] @ W[N,K]^T + bias)
// grid = (N/128, M/64), block = 256 (8 waves, each owns a 32x32 tile:
// 2x2 register blocking -> 4 loads feed 4 WMMAs per k-step).
// ---------------------------------------------------------------------------
__global__ void __launch_bounds__(256)
gemm_f32_wmma(const float* __restrict__ A, int lda,
              const float* __restrict__ W, int ldw,
              const float* __restrict__ bias,
              float* __restrict__ D, int ldd,
              int K, int act, int remap) {
  const int lane = threadIdx.x & 31;
  const int wid = threadIdx.x >> 5;
  const int lrow = lane & 15;
  const int khalf = lane >> 4;
  const int m0 = blockIdx.y * 64 + (wid >> 2) * 32;
  const int n0 = blockIdx.x * 128 + (wid & 3) * 32;

  const float* a0p = A + (size_t)(m0 + lrow) * lda + 2 * khalf;
  const float* a1p = a0p + (size_t)16 * lda;
  const float* w0p = W + (size_t)(n0 + lrow) * ldw + 2 * khalf;
  const float* w1p = w0p + (size_t)16 * ldw;

  v8f c00 = {}, c01 = {}, c10 = {}, c11 = {};
#pragma unroll 4
  for (int k = 0; k < K; k += 4) {
    v2f a0 = *(const v2f*)(a0p + k);
    v2f a1 = *(const v2f*)(a1p + k);
    v2f b0 = *(const v2f*)(w0p + k);
    v2f b1 = *(const v2f*)(w1p + k);
    c00 = wmma4(a0, b0, c00);
    c01 = wmma4(a0, b1, c01);
    c10 = wmma4(a1, b0, c10);
    c11 = wmma4(a1, b1, c11);
  }

  store_tile(c00, m0, n0, lrow, khalf, bias, D, ldd, act, remap);
  store_tile(c01, m0, n0 + 16, lrow, khalf, bias, D, ldd, act, remap);
  store_tile(c10, m0 + 16, n0, lrow, khalf, bias, D, ldd, act, remap);
  store_tile(c11, m0 + 16, n0 + 16, lrow, khalf, bias, D, ldd, act, remap);
}

// ---------------------------------------------------------------------------
// GRU layer-0 step: hnext = GRUcell(gi (precomputed, incl. bih), hprev)
// gh = hprev @ whh^T + bhh computed here; A-frag reused across 3 gate WMMAs.
// grid = 8 blocks * 8 waves = 64 tiles covering [32 x 512].
// ---------------------------------------------------------------------------
__global__ void __launch_bounds__(256)
gru_l0_step(const float* __restrict__ gi,     // [32, 1536] slice of GI0
            const float* __restrict__ hprev,  // [32, 512]
            const float* __restrict__ whh,    // [1536, 512]
            const float* __restrict__ bhh,    // [1536]
            float* __restrict__ hnext) {
  const int lane = threadIdx.x & 31;
  const int wid = threadIdx.x >> 5;
  const int tile = blockIdx.x * 8 + wid;      // 0..63
  const int n0 = (tile & 31) * 16;
  const int m0 = (tile >> 5) * 16;
  const int lrow = lane & 15, khalf = lane >> 4;

  const float* aptr = hprev + (m0 + lrow) * Hsz + 2 * khalf;
  const float* wr = whh + (size_t)(n0 + lrow) * Hsz + 2 * khalf;
  const float* wz = wr + (size_t)Hsz * Hsz;
  const float* wn = wz + (size_t)Hsz * Hsz;

  v8f accr = {}, accz = {}, accn = {};
#pragma unroll 4
  for (int k = 0; k < Hsz; k += 4) {
    v2f a = *(const v2f*)(aptr + k);
    v2f br = *(const v2f*)(wr + k);
    v2f bz = *(const v2f*)(wz + k);
    v2f bn = *(const v2f*)(wn + k);
    accr = wmma4(a, br, accr);
    accz = wmma4(a, bz, accz);
    accn = wmma4(a, bn, accn);
  }

  const int n = n0 + lrow;
  const float br_b = bhh[n], bz_b = bhh[n + Hsz], bn_b = bhh[n + 2 * Hsz];
  const int mb = m0 + 8 * khalf;
#pragma unroll
  for (int v = 0; v < 8; ++v) {
    int m = mb + v;                        // batch index
    const float* g = gi + m * (3 * Hsz) + n;
    float hp = hprev[m * Hsz + n];
    float r = sigmoidf_(g[0] + accr[v] + br_b);
    float z = sigmoidf_(g[Hsz] + accz[v] + bz_b);
    float nn = tanhf(g[2 * Hsz] + r * (accn[v] + bn_b));
    hnext[m * Hsz + n] = (1.0f - z) * nn + z * hp;
  }
}

// ---------------------------------------------------------------------------
// GRU layer-1 step: gi = x@wih^T+bih and gh = hprev@whh^T+bhh both computed
// here (6 WMMA accumulators per wave). Writes hnext and the per-step archive.
// ---------------------------------------------------------------------------
__global__ void __launch_bounds__(256)
gru_l1_step(const float* __restrict__ x,      // h0new [32, 512]
            const float* __restrict__ hprev,  // h1    [32, 512]
            const float* __restrict__ wih,    // [1536, 512] (layer 1)
            const float* __restrict__ whh,    // [1536, 512] (layer 1)
            const float* __restrict__ bih,    // [1536]
            const float* __restrict__ bhh,    // [1536]
            float* __restrict__ hnext,        // [32, 512]
            float* __restrict__ hstore) {     // H1all + t*B*H
  const int lane = threadIdx.x & 31;
  const int wid = threadIdx.x >> 5;
  const int tile = blockIdx.x * 8 + wid;
  const int n0 = (tile & 31) * 16;
  const int m0 = (tile >> 5) * 16;
  const int lrow = lane & 15, khalf = lane >> 4;

  const float* xptr = x + (m0 + lrow) * Hsz + 2 * khalf;
  const float* hptr = hprev + (m0 + lrow) * Hsz + 2 * khalf;
  const float* ir_w = wih + (size_t)(n0 + lrow) * Hsz + 2 * khalf;
  const float* iz_w = ir_w + (size_t)Hsz * Hsz;
  const float* in_w = iz_w + (size_t)Hsz * Hsz;
  const float* hr_w = whh + (size_t)(n0 + lrow) * Hsz + 2 * khalf;
  const float* hz_w = hr_w + (size_t)Hsz * Hsz;
  const float* hn_w = hz_w + (size_t)Hsz * Hsz;

  v8f ir = {}, iz = {}, in_ = {}, hr = {}, hz = {}, hn = {};
#pragma unroll 2
  for (int k = 0; k < Hsz; k += 4) {
    v2f ax = *(const v2f*)(xptr + k);
    v2f ah = *(const v2f*)(hptr + k);
    v2f b0 = *(const v2f*)(ir_w + k);
    v2f b1 = *(const v2f*)(iz_w + k);
    v2f b2 = *(const v2f*)(in_w + k);
    v2f b3 = *(const v2f*)(hr_w + k);
    v2f b4 = *(const v2f*)(hz_w + k);
    v2f b5 = *(const v2f*)(hn_w + k);
    ir = wmma4(ax, b0, ir);
    iz = wmma4(ax, b1, iz);
    in_ = wmma4(ax, b2, in_);
    hr = wmma4(ah, b3, hr);
    hz = wmma4(ah, b4, hz);
    hn = wmma4(ah, b5, hn);
  }

  const int n = n0 + lrow;
  const float bir = bih[n], biz = bih[n + Hsz], bin = bih[n + 2 * Hsz];
  const float bhr = bhh[n], bhz = bhh[n + Hsz], bhn = bhh[n + 2 * Hsz];
  const int mb = m0 + 8 * khalf;
#pragma unroll
  for (int v = 0; v < 8; ++v) {
    int m = mb + v;
    float hp = hprev[m * Hsz + n];
    float r = sigmoidf_(ir[v] + bir + hr[v] + bhr);
    float z = sigmoidf_(iz[v] + biz + hz[v] + bhz);
    float nn = tanhf(in_[v] + bin + r * (hn[v] + bhn));
    float h = (1.0f - z) * nn + z * hp;
    hnext[m * Hsz + n] = h;
    hstore[m * Hsz + n] = h;
  }
}

// ---------------------------------------------------------------------------
// Batched attention scores/softmax/mix. One block per (t,b). q already in
// A2[:, 512:1024] (written by the Q GEMM); mix written to A2[:, 0:512].
// attention weights written straight into d_out's attn_w region ([S,B,LEN]).
// ---------------------------------------------------------------------------
__global__ void __launch_bounds__(128)
attn_softmax_mix(const float* __restrict__ A2q, const float* __restrict__ ctx,
                 float* __restrict__ attnw, float* __restrict__ A2mix) {
  __shared__ float q[Hsz];
  __shared__ float sw[LENsz];
  __shared__ float red[LENsz];
  const int r = blockIdx.x;        // t*B + b
  const int b = r & 31;
  const int tid = threadIdx.x;     // 0..127

  const float* qsrc = A2q + (size_t)r * 1024 + Hsz;
  for (int i = tid; i < Hsz; i += 128) q[i] = qsrc[i];
  __syncthreads();

  const float* crow = ctx + ((size_t)b * LENsz + tid) * Hsz;
  float s = 0.0f;
#pragma unroll 4
  for (int k = 0; k < Hsz; k += 4) {
    s += q[k] * crow[k] + q[k + 1] * crow[k + 1] + q[k + 2] * crow[k + 2] +
         q[k + 3] * crow[k + 3];
  }

  red[tid] = s;
  __syncthreads();
  for (int off = 64; off > 0; off >>= 1) {
    if (tid < off) red[tid] = fmaxf(red[tid], red[tid + off]);
    __syncthreads();
  }
  const float mx = red[0];
  __syncthreads();
  const float e = __expf(s - mx);
  red[tid] = e;
  __syncthreads();
  for (int off = 64; off > 0; off >>= 1) {
    if (tid < off) red[tid] += red[tid + off];
    __syncthreads();
  }
  const float w = e / red[0];
  sw[tid] = w;
  attnw[(size_t)r * LENsz + tid] = w;
  __syncthreads();

  for (int h = tid; h < Hsz; h += 128) {
    const float* c0 = ctx + (size_t)b * LENsz * Hsz + h;
    float acc = 0.0f;
#pragma unroll 8
    for (int l = 0; l < LENsz; ++l) acc += sw[l] * c0[(size_t)l * Hsz];
    A2mix[(size_t)r * 1024 + h] = acc;
  }
}

// ---------------------------------------------------------------------------
extern "C" void kernel_launch(void* const* d_in, const int* in_sizes, int n_in,
                              void* d_out, int out_size, void* d_ws,
                              size_t ws_size, hipStream_t stream) {
  const int* text = (const int*)d_in[0];
  const float* dec_h = (const float*)d_in[1];   // [2,32,512]
  const float* ctx = (const float*)d_in[2];     // [32,128,512]
  const float* emb = (const float*)d_in[3];     // [32000,512]
  const float* wih = (const float*)d_in[4];     // [2,1536,512]
  const float* whh = (const float*)d_in[5];
  const float* bih = (const float*)d_in[6];     // [2,1536]
  const float* bhh = (const float*)d_in[7];
  const float* win = (const float*)d_in[8];     // [512,512]
  const float* wout = (const float*)d_in[9];    // [512,1024]
  const float* ow = (const float*)d_in[10];     // [32000,512]
  const float* ob = (const float*)d_in[11];     // [32000]
  float* out = (float*)d_out;

  const int SB = Ssz * Bsz;                     // 4096
  float* ws = (float*)d_ws;
  float* Xemb = ws;                             // 4096*512
  float* GI0 = Xemb + (size_t)SB * Hsz;         // 4096*1536
  float* h0a = GI0 + (size_t)SB * 3 * Hsz;
  float* h0b = h0a + Bsz * Hsz;
  float* h1a = h0b + Bsz * Hsz;
  float* h1b = h1a + Bsz * Hsz;
  float* H1all = h1b + Bsz * Hsz;               // 4096*512
  float* A2 = H1all + (size_t)SB * Hsz;         // 4096*1024 ([mix | q])
  float* Outs = A2 + (size_t)SB * 1024;         // 4096*512

  float* out_hfinal = out + (size_t)Bsz * Ssz * Vsz;
  float* out_attnw = out_hfinal + 2 * Bsz * Hsz;

  // initial hidden state (double-buffered across steps)
  hipMemcpyAsync(h0a, dec_h, (size_t)Bsz * Hsz * 4, hipMemcpyDeviceToDevice,
                 stream);
  hipMemcpyAsync(h1a, dec_h + Bsz * Hsz, (size_t)Bsz * Hsz * 4,
                 hipMemcpyDeviceToDevice, stream);

  embed_gather<<<SB, 256, 0, stream>>>(text, emb, Xemb);

  // GI0 = Xemb @ wih0^T + bih0   [4096,512]x[512,1536]
  gemm_f32_wmma<<<dim3((3 * Hsz) / 128, SB / 64), 256, 0, stream>>>(
      Xemb, Hsz, wih, Hsz, bih, GI0, 3 * Hsz, Hsz, 0, 0);

  // Sequential scan: 2 kernels per step (attention is off the critical path).
  float* h0in = h0a; float* h0out = h0b;
  float* h1in = h1a; float* h1out = h1b;
  for (int t = 0; t < Ssz; ++t) {
    gru_l0_step<<<8, 256, 0, stream>>>(GI0 + (size_t)t * Bsz * 3 * Hsz, h0in,
                                       whh, bhh, h0out);
    gru_l1_step<<<8, 256, 0, stream>>>(
        h0out, h1in, wih + (size_t)3 * Hsz * Hsz, whh + (size_t)3 * Hsz * Hsz,
        bih + 3 * Hsz, bhh + 3 * Hsz, h1out, H1all + (size_t)t * Bsz * Hsz);
    float* tmp = h0in; h0in = h0out; h0out = tmp;
    tmp = h1in; h1in = h1out; h1out = tmp;
  }
  // h_final = [h0(127), h1(127)]  (now in h*in after the final swap)
  hipMemcpyAsync(out_hfinal, h0in, (size_t)Bsz * Hsz * 4,
                 hipMemcpyDeviceToDevice, stream);
  hipMemcpyAsync(out_hfinal + Bsz * Hsz, h1in, (size_t)Bsz * Hsz * 4,
                 hipMemcpyDeviceToDevice, stream);

  // Batched attention: Q = H1all @ win^T -> A2[:,512:1024]
  gemm_f32_wmma<<<dim3(Hsz / 128, SB / 64), 256, 0, stream>>>(
      H1all, Hsz, win, Hsz, nullptr, A2 + Hsz, 1024, Hsz, 0, 0);
  attn_softmax_mix<<<SB, 128, 0, stream>>>(A2, ctx, out_attnw, A2);
  // Outs = tanh(A2 @ wout^T)   [4096,1024]x[1024,512]
  gemm_f32_wmma<<<dim3(Hsz / 128, SB / 64), 256, 0, stream>>>(
      A2, 1024, wout, 1024, nullptr, Outs, Hsz, 1024, 1, 0);

  // scores = Outs @ out_w^T + out_b, scattered to [B,S,V]
  gemm_f32_wmma<<<dim3(Vsz / 128, SB / 64), 256, 0, stream>>>(
      Outs, Hsz, ow, Hsz, ob, out, 0, Hsz, 0, 1);
}